// HashEncoder_38354057953505
// MI455X (gfx1250) — compile-verified
//
#include <hip/hip_runtime.h>
#include <hip/hip_bf16.h>
#include <cmath>

// Instant-NGP multires hash-grid encoder for MI455X (gfx1250, wave32).
//
// Roofline: ~0.5 GFLOP vs ~550 MB of data-dependent float2 gathers from a
// ~49 MB table that is fully L2-resident (192 MB L2). The kernel is bound by
// L2 random-gather throughput/latency, so the design maximizes independent
// global_load_b64 gathers in flight per wave (unrolled corners x unroll-2
// levels), keeps all control flow scalar (dense levels are a prefix), and
// streams the write-once output with non-temporal stores so it does not
// compete with the table for L2 capacity.
//
// WMMA/SWMMAC/TDM are inapplicable: the 8x2 blend operand differs per lane
// (random gather), so no wave-shared matrix operand exists, and TDM gather
// mode only supports <=16 SGPR-supplied row indices of a 2D tile.

#define KLEVELS 16
#define KFEAT   2
#define KHASH_MASK 0x7FFFFu   // every hashed level has size == 2^19 exactly

typedef float f32x2 __attribute__((ext_vector_type(2)));

struct LevelParams {
  float    scale[KLEVELS];   // float32(16 * ratio^l), ratio = 2^(7/15)
  unsigned res[KLEVELS];     // ceil(16 * ratio^l)
  unsigned off[KLEVELS];     // entry offset of level l in the table
  int      n_dense;          // levels [0, n_dense) are dense (res^3 <= 2^19)
};

__device__ __forceinline__ unsigned umin_(unsigned a, unsigned b) { return a < b ? a : b; }

__global__ void __launch_bounds__(256)
hash_encode_kernel(const float* __restrict__ coords,     // [N,3]
                   const f32x2* __restrict__ table,      // [total_entries] float2
                   float* __restrict__ out,              // [N, 32]
                   LevelParams P, int n)
{
  const int i = blockIdx.x * 256 + threadIdx.x;
  if (i >= n) return;

  const float x = coords[3 * i + 0];
  const float y = coords[3 * i + 1];
  const float z = coords[3 * i + 2];
  f32x2* __restrict__ op = (f32x2*)(out + (size_t)i * (KFEAT * KLEVELS));

  int l = 0;

  // ---- dense prefix: row-major grid index, corners clamped to res-1 ----
  for (; l < P.n_dense; ++l) {
    const float    s  = P.scale[l];
    const unsigned r  = P.res[l];
    const unsigned rm = r - 1u;
    const unsigned r2 = r * r;
    const unsigned base = P.off[l];

    const float px = x * s, py = y * s, pz = z * s;
    const float bx = floorf(px), by = floorf(py), bz = floorf(pz);
    const float fx = px - bx,  fy = py - by,  fz = pz - bz;
    const unsigned gx = (unsigned)bx, gy = (unsigned)by, gz = (unsigned)bz;

    // per-axis index contributions and weights (2 each)
    const unsigned xi[2] = { umin_(gx, rm),       umin_(gx + 1u, rm)      };
    const unsigned yi[2] = { umin_(gy, rm) * r,   umin_(gy + 1u, rm) * r  };
    const unsigned zi[2] = { umin_(gz, rm) * r2,  umin_(gz + 1u, rm) * r2 };
    const float wx[2] = { 1.f - fx, fx };
    const float wy[2] = { 1.f - fy, fy };
    const float wz[2] = { 1.f - fz, fz };

    float a0 = 0.f, a1 = 0.f;
    #pragma unroll
    for (int c = 0; c < 8; ++c) {          // 8 independent b64 gathers in flight
      const unsigned idx = xi[c & 1] + yi[(c >> 1) & 1] + zi[c >> 2];
      const f32x2 e = table[base + idx];
      const float w = wx[c & 1] * wy[(c >> 1) & 1] * wz[c >> 2];
      a0 += w * e.x;
      a1 += w * e.y;
    }
    f32x2 res; res.x = a0; res.y = a1;
    __builtin_nontemporal_store(res, op + l);   // NT: keep L2 for the table
  }

  // ---- hashed levels: xor of prime-multiplied coords, masked by 2^19-1 ----
  #pragma unroll 2                              // ~16 gathers in flight
  for (; l < KLEVELS; ++l) {
    const float    s    = P.scale[l];
    const unsigned base = P.off[l];

    const float px = x * s, py = y * s, pz = z * s;
    const float bx = floorf(px), by = floorf(py), bz = floorf(pz);
    const float fx = px - bx,  fy = py - by,  fz = pz - bz;
    const unsigned gx = (unsigned)bx, gy = (unsigned)by, gz = (unsigned)bz;

    // prime products per axis (uint32 wraparound matches the reference)
    const unsigned hx[2] = { gx,                  gx + 1u                  };
    const unsigned hy[2] = { gy * 2654435761u,    (gy + 1u) * 2654435761u  };
    const unsigned hz[2] = { gz * 805459861u,     (gz + 1u) * 805459861u   };
    const float wx[2] = { 1.f - fx, fx };
    const float wy[2] = { 1.f - fy, fy };
    const float wz[2] = { 1.f - fz, fz };

    float a0 = 0.f, a1 = 0.f;
    #pragma unroll
    for (int c = 0; c < 8; ++c) {
      const unsigned idx = (hx[c & 1] ^ hy[(c >> 1) & 1] ^ hz[c >> 2]) & KHASH_MASK;
      const f32x2 e = table[base + idx];
      const float w = wx[c & 1] * wy[(c >> 1) & 1] * wz[c >> 2];
      a0 += w * e.x;
      a1 += w * e.y;
    }
    f32x2 res; res.x = a0; res.y = a1;
    __builtin_nontemporal_store(res, op + l);
  }
}

extern "C" void kernel_launch(void* const* d_in, const int* in_sizes, int n_in,
                              void* d_out, int out_size, void* d_ws, size_t ws_size,
                              hipStream_t stream) {
  (void)n_in; (void)out_size; (void)d_ws; (void)ws_size;

  const float* coords = (const float*)d_in[0];   // [N,3] float32
  const f32x2* table  = (const f32x2*)d_in[1];   // [total, 2] float32 as float2
  // d_in[2] (hashmap_offsets) is recomputed host-side in float64, matching
  // the reference's _offsets() exactly (same libm exp2/pow/ceil calls).
  float* out = (float*)d_out;

  const int n = in_sizes[0] / 3;

  LevelParams P;
  const double ratio = std::exp2(7.0 / 15.0);    // exp2(log2(2048/16)/15)
  const unsigned long long maxp = 1ull << 19;
  unsigned long long off = 0;
  int n_dense = 0;
  for (int l = 0; l < KLEVELS; ++l) {
    const double s = 16.0 * std::pow(ratio, (double)l);
    const unsigned r = (unsigned)std::ceil(s);
    P.scale[l] = (float)s;
    P.res[l]   = r;
    P.off[l]   = (unsigned)off;
    const unsigned long long cube = (unsigned long long)r * r * r;
    if (cube <= maxp) n_dense = l + 1;           // dense levels form a prefix
    off += (cube < maxp) ? cube : maxp;
  }
  P.n_dense = n_dense;

  const int grid = (n + 255) / 256;
  hash_encode_kernel<<<grid, 256, 0, stream>>>(coords, table, out, P, n);
}